// XFeatSparseEncoder_10127532884219
// MI455X (gfx1250) — compile-verified
//
#include <hip/hip_runtime.h>
#include <hip/hip_bf16.h>
#include <cstdint>

// Problem constants (from reference)
#define B2V   64
#define BV    32
#define NV    1024
#define CV    129     // feature channels
#define CPAD  132     // C padded to multiple of 4 (zero pad) for sim GEMM
#define CINW  272     // cin row width: 129+129+3 extras, padded to mult of 16
#define HIDV  512
#define SIMLDS (16 * CPAD)   // one staged B tile: 16 rows x 132 f32 = 8448 B

typedef __attribute__((ext_vector_type(2))) float v2f;
typedef __attribute__((ext_vector_type(8))) float v8f;

// Monotonic float -> uint encoding (order-preserving), and inverse.
__device__ __forceinline__ unsigned int enc_f32(float v) {
  unsigned int b = __float_as_uint(v);
  return (b & 0x80000000u) ? ~b : (b | 0x80000000u);
}
__device__ __forceinline__ float dec_f32(unsigned int e) {
  unsigned int b = (e & 0x80000000u) ? (e & 0x7FFFFFFFu) : ~e;
  return __uint_as_float(b);
}

// ---------------------------------------------------------------------------
// Global min/max of heatmap (64*1024 values) -> mm[0]=min, mm[1]=max
// ---------------------------------------------------------------------------
__global__ void k_minmax(const float* __restrict__ hm, float* __restrict__ mm) {
  __shared__ float smn[1024];
  __shared__ float smx[1024];
  int t = threadIdx.x;
  float mn = 3.0e38f, mx = -3.0e38f;
  for (int i = t; i < B2V * NV; i += 1024) {
    float v = hm[i];
    mn = fminf(mn, v);
    mx = fmaxf(mx, v);
  }
  smn[t] = mn; smx[t] = mx;
  __syncthreads();
  for (int s = 512; s > 0; s >>= 1) {
    if (t < s) {
      smn[t] = fminf(smn[t], smn[t + s]);
      smx[t] = fmaxf(smx[t], smx[t + s]);
    }
    __syncthreads();
  }
  if (t == 0) { mm[0] = smn[0]; mm[1] = smx[0]; }
}

// ---------------------------------------------------------------------------
// t_scale[b] = sum_c2 mean_hw(full[2b + c2/129][c2%129]) * Ws[c2] + bs
// ---------------------------------------------------------------------------
__global__ void k_tscale(const float* __restrict__ feats, const float* __restrict__ kl,
                         const float* __restrict__ Ws, const float* __restrict__ bs,
                         float* __restrict__ out) {
  int b = blockIdx.x;  // 0..31
  __shared__ float part[256];
  float acc = 0.f;
  for (int c2 = threadIdx.x; c2 < 2 * CV; c2 += 256) {
    int b2 = b * 2 + c2 / CV;
    int c  = c2 % CV;
    const float* src = (c < 64) ? (feats + ((size_t)b2 * 64 + c) * NV)
                                : (kl    + ((size_t)b2 * 65 + (c - 64)) * NV);
    float s = 0.f;
    for (int i = 0; i < NV; ++i) s += src[i];
    acc += (s * (1.0f / 1024.0f)) * Ws[c2];
  }
  part[threadIdx.x] = acc;
  __syncthreads();
  for (int s = 128; s > 0; s >>= 1) {
    if (threadIdx.x < s) part[threadIdx.x] += part[threadIdx.x + s];
    __syncthreads();
  }
  if (threadIdx.x == 0) out[b] = part[0] + bs[0];
}

// ---------------------------------------------------------------------------
// Stable descending sort via rank counting (matches jax.lax.top_k tie rule:
// equal values -> lower index first).
// ---------------------------------------------------------------------------
__global__ void k_sort(const float* __restrict__ hm, float* __restrict__ ts,
                       int* __restrict__ tidx) {
  int b = blockIdx.x;  // 0..63
  __shared__ float v[NV];
  const float* row = hm + (size_t)b * NV;
  for (int i = threadIdx.x; i < NV; i += blockDim.x) v[i] = row[i];
  __syncthreads();
  for (int i = threadIdx.x; i < NV; i += blockDim.x) {
    float vi = v[i];
    int r = 0;
    for (int j = 0; j < NV; ++j) {
      float vj = v[j];
      r += (vj > vi) || (vj == vi && j < i);
    }
    ts[(size_t)b * NV + r]   = vi;
    tidx[(size_t)b * NV + r] = i;
  }
}

// ---------------------------------------------------------------------------
// Gather sorted features. Even b2 -> cin cols [0..128] (f0) + zero pad
// [129..131]; odd b2 -> f1 (padded to 132). Even also writes k0 coords.
// ---------------------------------------------------------------------------
__global__ void k_gather(const float* __restrict__ feats, const float* __restrict__ kl,
                         const int* __restrict__ tidx, float* __restrict__ cin,
                         float* __restrict__ f1, float* __restrict__ k0b) {
  int b2 = blockIdx.x;
  int n  = blockIdx.y * 256 + threadIdx.x;
  int b  = b2 >> 1;
  int pr = b2 & 1;
  int idx = tidx[(size_t)b2 * NV + n];
  float* dst = pr ? (f1 + ((size_t)b * NV + n) * CPAD)
                  : (cin + ((size_t)b * NV + n) * CINW);
  const float* fb = feats + (size_t)b2 * 64 * NV;
  const float* kb = kl    + (size_t)b2 * 65 * NV;
  for (int c = 0; c < 64; ++c) dst[c]      = fb[(size_t)c * NV + idx];
  for (int c = 0; c < 65; ++c) dst[64 + c] = kb[(size_t)c * NV + idx];
  dst[129] = 0.f; dst[130] = 0.f; dst[131] = 0.f;   // zero K-pad for sim GEMM
  if (!pr) {
    k0b[((size_t)b * NV + n) * 2 + 0] = (float)(idx & 31);
    k0b[((size_t)b * NV + n) * 2 + 1] = (float)(idx >> 5);
  }
}

__global__ void k_init(unsigned long long* __restrict__ rowbest,
                       unsigned long long* __restrict__ colbest) {
  int i = blockIdx.x * 256 + threadIdx.x;   // 0..32767
  rowbest[i] = 0ULL;
  colbest[i] = 0ULL;
}

// ---------------------------------------------------------------------------
// Pack weight matrix W[K x N] into pair-interleaved, zero-padded layout:
// Wp[p * N + n] = { W[2p][n], W[2p+1][n] }, p in [0, P), rows >= K are zero.
// Gives single aligned float2 B-operand loads in the GEMM (WMMA B layout
// wants {W[ka][n], W[ka+1][n]} in one lane).
// ---------------------------------------------------------------------------
__global__ void k_wpack(const float* __restrict__ W, int K, int N, int P,
                        v2f* __restrict__ out) {
  int i = blockIdx.x * 256 + threadIdx.x;
  if (i >= P * N) return;
  int p = i / N, n = i - p * N;
  float w0 = (2 * p     < K) ? W[(size_t)(2 * p) * N + n]     : 0.f;
  float w1 = (2 * p + 1 < K) ? W[(size_t)(2 * p + 1) * N + n] : 0.f;
  v2f o = {w0, w1};
  out[i] = o;
}

// ---------------------------------------------------------------------------
// Similarity GEMM: ms[b] = f0 @ f1^T / sqrt(129) via V_WMMA_F32_16X16X4_F32.
//
// A operands (33 float2) live in registers for the whole kernel. B tiles are
// contiguous 8448-byte blocks of f1, cooperatively staged into LDS with
// coalesced b128 copies and DOUBLE-BUFFERED: while the 8 waves run the
// 33-WMMA chain against tile nt (conflict-free ds_load_b64: lane l16 hits
// bank 4*l16 + 2*lhalf), the block fetches tile nt+1. This cuts global B
// traffic 8x versus per-wave streaming and replaces long loadcnt stalls in
// the MMA chain with short LDS reads. Two interleaved accumulators break the
// serial D->C chain.
//
// Row/col argmaxes kept as packed u64 keys:
// key = enc(value)<<32 | (0xFFFFFFFF - index)  => max key == argmax with
// lowest-index tie-breaking, exactly like jnp.argmax.
// ---------------------------------------------------------------------------
__global__ void k_sim(const float* __restrict__ cin, const float* __restrict__ f1,
                      unsigned long long* __restrict__ rowbest,
                      unsigned long long* __restrict__ colbest) {
  int b      = blockIdx.x;   // 0..31
  int mstrip = blockIdx.y;   // 0..7 -> 128 rows each
  int wave = threadIdx.x >> 5;
  int lane = threadIdx.x & 31;
  int l16 = lane & 15, lhalf = lane >> 4;
  int m0 = mstrip * 128 + wave * 16;

  __shared__ __align__(16) float Bs[2][SIMLDS];
  __shared__ unsigned long long rb[128];
  for (int i = threadIdx.x; i < 128; i += 256) rb[i] = 0ULL;

  // A: rows m0..m0+15 of f0 (= cin cols 0..131, row stride CINW)
  const float* Arow  = cin + ((size_t)b * NV + m0 + l16) * CINW;
  const float* Bbase = f1 + (size_t)b * NV * CPAD;
  const float invs = 0.08804509063256238f;  // 1/sqrt(129)

  v2f av[33];
#pragma unroll
  for (int kk = 0; kk < 33; ++kk)
    av[kk] = *(const v2f*)(Arow + kk * 4 + 2 * lhalf);

  // Prologue: stage tile 0
  {
    const float4* src = (const float4*)Bbase;
    float4* dstp = (float4*)(&Bs[0][0]);
    for (int i = threadIdx.x; i < SIMLDS / 4; i += 256) dstp[i] = src[i];
  }
  __syncthreads();

  int buf = 0;
  for (int nt = 0; nt < 64; ++nt) {
    // Prefetch tile nt+1 into the other buffer while computing on this one.
    if (nt + 1 < 64) {
      const float4* src = (const float4*)(Bbase + (size_t)(nt + 1) * 16 * CPAD);
      float4* dstp = (float4*)(&Bs[buf ^ 1][0]);
      for (int i = threadIdx.x; i < SIMLDS / 4; i += 256) dstp[i] = src[i];
    }

    const float* Bt = &Bs[buf][0] + (size_t)l16 * CPAD + 2 * lhalf;
    v2f bv[33];
#pragma unroll
    for (int kk = 0; kk < 33; ++kk)
      bv[kk] = *(const v2f*)(Bt + kk * 4);

    v8f acc0 = {0.f, 0.f, 0.f, 0.f, 0.f, 0.f, 0.f, 0.f};
    v8f acc1 = {0.f, 0.f, 0.f, 0.f, 0.f, 0.f, 0.f, 0.f};
#pragma unroll
    for (int kk = 0; kk < 32; kk += 2) {
      acc0 = __builtin_amdgcn_wmma_f32_16x16x4_f32(false, av[kk], false, bv[kk],
                                                   (short)0, acc0, false, false);
      acc1 = __builtin_amdgcn_wmma_f32_16x16x4_f32(false, av[kk + 1], false, bv[kk + 1],
                                                   (short)0, acc1, false, false);
    }
    acc0 = __builtin_amdgcn_wmma_f32_16x16x4_f32(false, av[32], false, bv[32],
                                                 (short)0, acc0, false, false);

    // D layout: element j = (m0 + j + 8*lhalf, n)
    int n = nt * 16 + l16;
    unsigned long long cmax = 0ULL;
#pragma unroll
    for (int j = 0; j < 8; ++j) {
      float val = (acc0[j] + acc1[j]) * invs;
      unsigned int e = enc_f32(val);
      int m = m0 + j + 8 * lhalf;
      unsigned long long rkey =
          ((unsigned long long)e << 32) | (unsigned long long)(0xFFFFFFFFu - (unsigned)n);
      atomicMax(&rb[m - mstrip * 128], rkey);
      unsigned long long ckey =
          ((unsigned long long)e << 32) | (unsigned long long)(0xFFFFFFFFu - (unsigned)m);
      if (ckey > cmax) cmax = ckey;
    }
    atomicMax(&colbest[(size_t)b * NV + n], cmax);

    __syncthreads();
    buf ^= 1;
  }
  for (int i = threadIdx.x; i < 128; i += 256)
    rowbest[(size_t)b * NV + mstrip * 128 + i] = rb[i];
}

// ---------------------------------------------------------------------------
// Mutual matching; fills cin cols 129..271 (f1_sel, s0, s1_sel, add_in, pad),
// valid flags, and kpts1 output. Pad cols are explicitly zeroed so no
// uninitialized value (potential NaN on first call) enters a WMMA.
// ---------------------------------------------------------------------------
__global__ void k_match(const unsigned long long* __restrict__ rowbest,
                        const unsigned long long* __restrict__ colbest,
                        const float* __restrict__ ts, const int* __restrict__ tidx,
                        const float* __restrict__ f1, const float* __restrict__ mm,
                        float* __restrict__ cin, float* __restrict__ validf,
                        float* __restrict__ out_kpts1) {
  int b = blockIdx.x;
  int m = blockIdx.y * 256 + threadIdx.x;
  unsigned long long rk = rowbest[(size_t)b * NV + m];
  int nn12 = (int)(0xFFFFFFFFu - (unsigned int)(rk & 0xFFFFFFFFu));
  unsigned long long ck = colbest[(size_t)b * NV + nn12];
  int nn21 = (int)(0xFFFFFFFFu - (unsigned int)(ck & 0xFFFFFFFFu));
  bool mutual = (nn21 == m);
  float vf = mutual ? 1.0f : 0.0f;
  int idx = mutual ? nn12 : 0;  // clip(-1,0,N-1) -> 0, masked by vf anyway
  float addv = mutual ? dec_f32((unsigned int)(rk >> 32)) : 0.0f;

  float gmin = mm[0], rng = mm[1] - mm[0];
  float s0  = (ts[(size_t)(2 * b) * NV + m] - gmin) / rng;
  float s1s = ((ts[(size_t)(2 * b + 1) * NV + idx] - gmin) / rng) * vf;

  float* crow = cin + ((size_t)b * NV + m) * CINW;
  const float* f1row = f1 + ((size_t)b * NV + idx) * CPAD;
  for (int c = 0; c < CV; ++c) crow[CV + c] = f1row[c] * vf;
  crow[258] = s0;
  crow[259] = s1s;
  crow[260] = addv;
  for (int c = 261; c < CINW; ++c) crow[c] = 0.f;
  validf[(size_t)b * NV + m] = vf;

  int ki = tidx[(size_t)(2 * b + 1) * NV + idx];
  out_kpts1[((size_t)b * NV + m) * 2 + 0] = (float)(ki & 31) * vf;
  out_kpts1[((size_t)b * NV + m) * 2 + 1] = (float)(ki >> 5) * vf;
}

// ---------------------------------------------------------------------------
// Generic f32 WMMA GEMM: C[M x N] = act(A[M x Kpad](lda) @ Wp + bias)
// Wp is pair-interleaved zero-padded (see k_wpack), so all operands are
// unguarded aligned float2 loads. Block = 256 thr (8 waves); each wave does a
// 16-row x 64-col tile (4 independent accumulator chains). K is processed in
// 16-element chunks: 20 loads issued, then 16 WMMAs, so loads batch and
// drain while the matrix pipe works.
// ---------------------------------------------------------------------------
__global__ void k_gemm(const float* __restrict__ A, int lda, int Kpad,
                       const v2f* __restrict__ Wp, int N,
                       const float* __restrict__ bias,
                       float* __restrict__ C, int ldc, int relu) {
  int wave = threadIdx.x >> 5;
  int lane = threadIdx.x & 31;
  int l16 = lane & 15, lhalf = lane >> 4;
  int m0 = blockIdx.x * 128 + wave * 16;
  int n0 = blockIdx.y * 64;
  const float* Arow = A + (size_t)(m0 + l16) * lda;

  v8f acc[4];
#pragma unroll
  for (int t = 0; t < 4; ++t) acc[t] = (v8f){0.f, 0.f, 0.f, 0.f, 0.f, 0.f, 0.f, 0.f};

  for (int k0 = 0; k0 < Kpad; k0 += 16) {
    v2f av[4];
    v2f bv[4][4];
#pragma unroll
    for (int s = 0; s < 4; ++s) {
      int ka = k0 + 4 * s + 2 * lhalf;
      av[s] = *(const v2f*)(Arow + ka);
      const v2f* Wrow = Wp + (size_t)(k0 / 2 + 2 * s + lhalf) * N;
#pragma unroll
      for (int t = 0; t < 4; ++t) bv[s][t] = Wrow[n0 + t * 16 + l16];
    }
#pragma unroll
    for (int s = 0; s < 4; ++s) {
#pragma unroll
      for (int t = 0; t < 4; ++t)
        acc[t] = __builtin_amdgcn_wmma_f32_16x16x4_f32(false, av[s], false, bv[s][t],
                                                       (short)0, acc[t], false, false);
    }
  }
#pragma unroll
  for (int t = 0; t < 4; ++t) {
    int n = n0 + t * 16 + l16;
    float bvv = bias[n];
#pragma unroll
    for (int j = 0; j < 8; ++j) {
      int m = m0 + j + 8 * lhalf;
      float v = acc[t][j] + bvv;
      if (relu) v = fmaxf(v, 0.f);
      C[(size_t)m * ldc + n] = v;
    }
  }
}

// conf = sigmoid(h2 . Wc2 + bc2) * valid
__global__ void k_conf(const float* __restrict__ h2, const float* __restrict__ Wc2,
                       const float* __restrict__ bc2, const float* __restrict__ validf,
                       float* __restrict__ out_conf) {
  int i = blockIdx.x * 256 + threadIdx.x;  // 0..32767
  const float* row = h2 + (size_t)i * 64;
  float s = bc2[0];
  for (int j = 0; j < 64; ++j) s += row[j] * Wc2[j];
  float c = 1.0f / (1.0f + expf(-s));
  out_conf[i] = c * validf[i];
}

// 8x8 softmax centroid (temp 3.0) + k0
__global__ void k_subpix(const float* __restrict__ offb, const float* __restrict__ k0b,
                         float* __restrict__ out_kpts0) {
  int i = blockIdx.x * 256 + threadIdx.x;  // 0..32767
  const float* row = offb + (size_t)i * 64;
  float mx = -3.0e38f;
  for (int j = 0; j < 64; ++j) mx = fmaxf(mx, row[j]);
  float sum = 0.f, cx = 0.f, cy = 0.f;
  for (int j = 0; j < 64; ++j) {
    float e = expf(3.0f * (row[j] - mx));
    sum += e;
    cx += e * (float)((j & 7) - 4);
    cy += e * (float)((j >> 3) - 4);
  }
  cx /= sum; cy /= sum;
  out_kpts0[(size_t)i * 2 + 0] = cx + k0b[(size_t)i * 2 + 0];
  out_kpts0[(size_t)i * 2 + 1] = cy + k0b[(size_t)i * 2 + 1];
}

// ---------------------------------------------------------------------------
extern "C" void kernel_launch(void* const* d_in, const int* in_sizes, int n_in,
                              void* d_out, int out_size, void* d_ws, size_t ws_size,
                              hipStream_t stream) {
  (void)in_sizes; (void)n_in; (void)out_size; (void)ws_size;
  const float* feats = (const float*)d_in[0];
  const float* kl    = (const float*)d_in[1];
  const float* hm    = (const float*)d_in[2];
  // d_in[3] = intrinsics: dead in reference outputs
  const float* Wc1 = (const float*)d_in[4];
  const float* bc1 = (const float*)d_in[5];
  const float* Wc2 = (const float*)d_in[6];
  const float* bc2 = (const float*)d_in[7];
  const float* Wf1 = (const float*)d_in[8];
  const float* bf1 = (const float*)d_in[9];
  const float* Wf2 = (const float*)d_in[10];
  const float* bf2 = (const float*)d_in[11];
  const float* Ws  = (const float*)d_in[12];
  const float* bs  = (const float*)d_in[13];

  float* out = (float*)d_out;
  float* out_kpts0 = out;                 // (32,1024,2)
  float* out_kpts1 = out + 65536;         // (32,1024,2)
  float* out_conf  = out + 131072;        // (32,1024,1)
  float* out_tsc   = out + 163840;        // (32,1)

  char* ws = (char*)d_ws;
  float* ts     = (float*)(ws + 0);                      //  64*1024 f32
  int*   tidx   = (int*)  (ws + 262144);                 //  64*1024 i32
  float* f1     = (float*)(ws + 524288);                 //  32*1024*132 f32
  float* k0b    = (float*)(ws + 17825792);               //  32*1024*2 f32
  unsigned long long* rowbest = (unsigned long long*)(ws + 18087936);  // 32*1024 u64
  unsigned long long* colbest = (unsigned long long*)(ws + 18350080);  // 32*1024 u64
  float* validf = (float*)(ws + 18612224);               //  32*1024 f32
  float* mm     = (float*)(ws + 18743296);               //  min,max
  float* cin    = (float*)(ws + 18743552);               //  32*1024*272 f32
  v2f*   Wf1p   = (v2f*)  (ws + 54395136);               //  136*512 v2f
  v2f*   Wf2p   = (v2f*)  (ws + 54952192);               //  256*64  v2f
  v2f*   Wc1p   = (v2f*)  (ws + 55083264);               //  136*64  v2f
  float* h1     = (float*)(ws + 55152896);               //  4096*512 f32 (chunk)
  float* offb   = (float*)(ws + 63541504);               //  32768*64 f32
  float* h2     = (float*)(ws + 71930112);               //  32768*64 f32

  k_minmax<<<1, 1024, 0, stream>>>(hm, mm);
  k_tscale<<<32, 256, 0, stream>>>(feats, kl, Ws, bs, out_tsc);
  k_sort<<<64, 256, 0, stream>>>(hm, ts, tidx);
  k_gather<<<dim3(64, 4), 256, 0, stream>>>(feats, kl, tidx, cin, f1, k0b);
  k_init<<<128, 256, 0, stream>>>(rowbest, colbest);

  // Pack weights: Kpad(258)=272 -> P=136; Kpad(512)=512 -> P=256; Kpad(261)=272
  k_wpack<<<(136 * 512 + 255) / 256, 256, 0, stream>>>(Wf1, 258, 512, 136, Wf1p);
  k_wpack<<<(256 * 64 + 255) / 256, 256, 0, stream>>>(Wf2, 512, 64, 256, Wf2p);
  k_wpack<<<(136 * 64 + 255) / 256, 256, 0, stream>>>(Wc1, 261, 64, 136, Wc1p);

  k_sim<<<dim3(32, 8), 256, 0, stream>>>(cin, f1, rowbest, colbest);
  k_match<<<dim3(32, 4), 256, 0, stream>>>(rowbest, colbest, ts, tidx, f1, mm,
                                           cin, validf, out_kpts1);

  // MLP offsets: relu(fin @ Wf1) @ Wf2, M chunked 8x to bound workspace
  const int CHUNK = 4096;
  for (int c = 0; c < (BV * NV) / CHUNK; ++c) {
    const float* Achunk = cin + (size_t)c * CHUNK * CINW;
    k_gemm<<<dim3(CHUNK / 128, HIDV / 64), 256, 0, stream>>>(
        Achunk, CINW, 272, Wf1p, HIDV, bf1, h1, HIDV, 1);
    k_gemm<<<dim3(CHUNK / 128, 1), 256, 0, stream>>>(
        h1, HIDV, 512, Wf2p, 64, bf2, offb + (size_t)c * CHUNK * 64, 64, 0);
  }
  // conf hidden: relu(cin @ Wc1)
  k_gemm<<<dim3((BV * NV) / 128, 1), 256, 0, stream>>>(
      cin, CINW, 272, Wc1p, 64, bc1, h2, 64, 1);

  k_conf<<<128, 256, 0, stream>>>(h2, Wc2, bc2, validf, out_conf);
  k_subpix<<<128, 256, 0, stream>>>(offb, k0b, out_kpts0);
}